// GATLayer_36558761623739
// MI455X (gfx1250) — compile-verified
//
#include <hip/hip_runtime.h>
#include <math.h>

#define K_DIM 128
#define D_DIM 64

typedef __attribute__((ext_vector_type(2))) float v2f;
typedef __attribute__((ext_vector_type(8))) float v8f;

// ---------------------------------------------------------------------------
// Kernel 1: z = h @ W^T via V_WMMA_F32_16X16X4_F32.
// Block = 128 threads (4 waves). Each block -> 16 rows of z; each wave -> one
// 16x16 column tile (64 cols = 4 tiles). 32 WMMA ops per wave (K=128, 4/step).
// ---------------------------------------------------------------------------
__global__ __launch_bounds__(128) void gat_proj_wmma(
    const float* __restrict__ h, const float* __restrict__ W,
    float* __restrict__ z)
{
  const int lane = threadIdx.x & 31;
  const int wave = threadIdx.x >> 5;         // 0..3 -> column tile
  const int l15  = lane & 15;
  const int kb   = (lane >> 4) << 1;         // lanes 0-15: K base 0, 16-31: 2
  const int row_base = blockIdx.x << 4;
  const int col_base = wave << 4;

  const float* __restrict__ hrow = h + (size_t)(row_base + l15) * K_DIM;
  const float* __restrict__ wrow = W + (size_t)(col_base + l15) * K_DIM;

  v8f acc = {};
#pragma unroll
  for (int k = 0; k < K_DIM; k += 4) {
    v2f a, b;
    // A (16x4 of h): lane l15 = row M, components = K kb, kb+1
    a.x = hrow[k + kb];
    a.y = hrow[k + kb + 1];
    // B (4x16 of W^T): B[k][n] = W[n][k]; lane l15 = col N
    b.x = wrow[k + kb];
    b.y = wrow[k + kb + 1];
    acc = __builtin_amdgcn_wmma_f32_16x16x4_f32(
        /*neg_a=*/false, a, /*neg_b=*/false, b,
        /*c_mod=*/(short)0, acc, /*reuse_a=*/false, /*reuse_b=*/false);
  }

  const int m0 = (lane >> 4) << 3;           // lanes 16-31 hold M+8
#pragma unroll
  for (int v = 0; v < 8; ++v) {
    z[(size_t)(row_base + m0 + v) * D_DIM + col_base + l15] = acc[v];
  }
}

// ---------------------------------------------------------------------------
// Kernel 2: per-node attention scalars s1[n] = z[n].a[:64], s2[n] = z[n].a[64:]
// (algebraically replaces the [E,64] gathers of the reference)
// ---------------------------------------------------------------------------
__global__ __launch_bounds__(256) void gat_scores(
    const float* __restrict__ z, const float* __restrict__ a,
    float* __restrict__ s1, float* __restrict__ s2, int nN)
{
  int i = blockIdx.x * 256 + threadIdx.x;
  if (i >= nN) return;
  const float* zr = z + (size_t)i * D_DIM;
  float acc1 = 0.f, acc2 = 0.f;
#pragma unroll
  for (int d = 0; d < D_DIM; ++d) {
    float zv = zr[d];
    acc1 += zv * a[d];
    acc2 += zv * a[D_DIM + d];
  }
  s1[i] = acc1;
  s2[i] = acc2;
}

// ---------------------------------------------------------------------------
// Kernel 3: init out = 0, m = -inf (bits), s = 0 (harness poisons buffers)
// ---------------------------------------------------------------------------
__global__ __launch_bounds__(256) void gat_init(
    int* __restrict__ mbits, float* __restrict__ ssum,
    float* __restrict__ out, int nN)
{
  size_t i = (size_t)blockIdx.x * 256 + threadIdx.x;
  if (i < (size_t)nN * D_DIM) out[i] = 0.f;
  if (i < (size_t)nN) {
    mbits[i] = (int)0xFF800000u;  // -inf
    ssum[i]  = 0.f;
  }
}

__device__ __forceinline__ float leaky01(float v) {
  return v >= 0.f ? v : 0.01f * v;
}

// Ordered-float atomic max trick: signed max for >=0, unsigned min for <0.
__device__ __forceinline__ void atomicMaxFloatBits(int* addr, float v) {
  if (v >= 0.f)
    atomicMax(addr, __float_as_int(v));
  else
    atomicMin((unsigned int*)addr, __float_as_uint(v));
}

// ---------------------------------------------------------------------------
// Kernel 4: segment max of edge logits into m[dst]
// ---------------------------------------------------------------------------
__global__ __launch_bounds__(256) void gat_edge_max(
    const float* __restrict__ s1, const float* __restrict__ s2,
    const int* __restrict__ src, const int* __restrict__ dst,
    int* __restrict__ mbits, int nE)
{
  int e = blockIdx.x * 256 + threadIdx.x;
  if (e >= nE) return;
  int d = dst[e];
  float v = leaky01(s1[src[e]] + s2[d]);
  atomicMaxFloatBits(&mbits[d], v);
}

// ---------------------------------------------------------------------------
// Kernel 5: segment sum of exp(e - m[dst]) into ssum[dst]
// ---------------------------------------------------------------------------
__global__ __launch_bounds__(256) void gat_edge_sum(
    const float* __restrict__ s1, const float* __restrict__ s2,
    const int* __restrict__ src, const int* __restrict__ dst,
    const int* __restrict__ mbits, float* __restrict__ ssum, int nE)
{
  int e = blockIdx.x * 256 + threadIdx.x;
  if (e >= nE) return;
  int d = dst[e];
  float v = leaky01(s1[src[e]] + s2[d]);
  float ex = __expf(v - __int_as_float(mbits[d]));
  unsafeAtomicAdd(&ssum[d], ex);   // native GLOBAL_ATOMIC_ADD_F32
}

// ---------------------------------------------------------------------------
// Kernel 6: out[dst] += alpha * z[src]; 16 threads per edge, float4 per thread
// ---------------------------------------------------------------------------
__global__ __launch_bounds__(256) void gat_scatter(
    const float* __restrict__ z,
    const float* __restrict__ s1, const float* __restrict__ s2,
    const int* __restrict__ src, const int* __restrict__ dst,
    const int* __restrict__ mbits, const float* __restrict__ ssum,
    float* __restrict__ out, int nE)
{
  int t = blockIdx.x * 256 + threadIdx.x;
  int e = t >> 4;
  int g = t & 15;                  // which group of 4 columns
  if (e >= nE) return;
  int s = src[e], d = dst[e];
  float v = leaky01(s1[s] + s2[d]);
  float alpha = __expf(v - __int_as_float(mbits[d])) / ssum[d];

  const float4 zv = *reinterpret_cast<const float4*>(z + (size_t)s * D_DIM + g * 4);
  float* o = out + (size_t)d * D_DIM + g * 4;
  unsafeAtomicAdd(o + 0, alpha * zv.x);
  unsafeAtomicAdd(o + 1, alpha * zv.y);
  unsafeAtomicAdd(o + 2, alpha * zv.z);
  unsafeAtomicAdd(o + 3, alpha * zv.w);
}

// ---------------------------------------------------------------------------
extern "C" void kernel_launch(void* const* d_in, const int* in_sizes, int n_in,
                              void* d_out, int out_size, void* d_ws, size_t ws_size,
                              hipStream_t stream)
{
  const float* h   = (const float*)d_in[0];   // [N,128]
  const float* W   = (const float*)d_in[1];   // [64,128]
  const float* a   = (const float*)d_in[2];   // [1,128]
  const int*   src = (const int*)d_in[3];     // [E]
  const int*   dst = (const int*)d_in[4];     // [E]
  float* out = (float*)d_out;                 // [N,64]

  const int nN = in_sizes[0] / K_DIM;         // 100000 (multiple of 16)
  const int nE = in_sizes[3];                 // 1250000

  // Workspace layout: z[N*64] | s1[N] | s2[N] | mbits[N] | ssum[N]  (~27.2 MB)
  float* z    = (float*)d_ws;
  float* s1   = z + (size_t)nN * D_DIM;
  float* s2   = s1 + nN;
  int*   mb   = (int*)(s2 + nN);
  float* ssum = (float*)(mb + nN);

  gat_proj_wmma<<<nN / 16, 128, 0, stream>>>(h, W, z);
  gat_scores  <<<(nN + 255) / 256, 256, 0, stream>>>(z, a, s1, s2, nN);
  gat_init    <<<((size_t)nN * D_DIM + 255) / 256, 256, 0, stream>>>(mb, ssum, out, nN);
  gat_edge_max<<<(nE + 255) / 256, 256, 0, stream>>>(s1, s2, src, dst, mb, nE);
  gat_edge_sum<<<(nE + 255) / 256, 256, 0, stream>>>(s1, s2, src, dst, mb, ssum, nE);
  gat_scatter <<<((size_t)nE * 16 + 255) / 256, 256, 0, stream>>>(z, s1, s2, src, dst, mb, ssum, out, nE);
}